// RConv_88192858456461
// MI455X (gfx1250) — compile-verified
//
#include <hip/hip_runtime.h>
#include <hip/hip_bf16.h>

#define N_NODES 10000
#define N_EDGES 100000
#define IN_F 64
#define OUT_F 64
#define N_REL 10

typedef __attribute__((ext_vector_type(2))) float v2f;
typedef __attribute__((ext_vector_type(8))) float v8f;

// ---------------------------------------------------------------- utilities
__global__ __launch_bounds__(256) void zero_kernel(float* __restrict__ p, int n) {
    int t = blockIdx.x * blockDim.x + threadIdx.x;
    if (t < n) p[t] = 0.0f;
}

// ------------------------------------------------------------- degree count
__global__ __launch_bounds__(256) void degree_kernel(const int* __restrict__ src,
                                                     const int* __restrict__ dst,
                                                     float* __restrict__ deg_out,
                                                     float* __restrict__ deg_in,
                                                     int n_edges) {
    int e = blockIdx.x * blockDim.x + threadIdx.x;
    if (e >= n_edges) return;
    atomicAdd(&deg_out[src[e]], 1.0f);
    atomicAdd(&deg_in[dst[e]], 1.0f);
}

// --------------------------------------------- feat_n = feat * out_deg^-1/2
__global__ __launch_bounds__(256) void prenorm_kernel(const float* __restrict__ feat,
                                                      const float* __restrict__ deg_out,
                                                      float* __restrict__ feat_n,
                                                      int n_elems) {
    int t = blockIdx.x * blockDim.x + threadIdx.x;
    if (t >= n_elems) return;
    int node = t >> 6;  // /64
    float s = rsqrtf(fmaxf(deg_out[node], 1.0f));
    feat_n[t] = feat[t] * s;
}

// ------------------------------------------------------------ WMMA GEMM
// H[r, m, o] = sum_i feat_n[m, i] * W_r[o, i]   (W_r row-major [out, in])
// One wave computes a 16-node x 64-out tile. 8 waves / block => 128 nodes.
// A is 16x4 fp32 (2 VGPR/lane), B is 4x16 fp32 (W^T tile), C/D 16x16 (v8f).
#define SW_STRIDE 68  // pad 64->68 floats: avoid LDS bank conflicts, keep 8B align

__global__ __launch_bounds__(256) void rgcn_gemm_kernel(const float* __restrict__ feat_n,
                                                        const float* __restrict__ emb,
                                                        float* __restrict__ H) {
    __shared__ float sW[OUT_F * SW_STRIDE];  // W_r, padded rows

    const int r = blockIdx.y;
    // cooperatively stage this relation's 64x64 weight into LDS
    const float* wr = emb + (size_t)r * (OUT_F * IN_F);
    for (int idx = threadIdx.x; idx < OUT_F * IN_F; idx += blockDim.x) {
        int o = idx >> 6, i = idx & 63;
        sW[o * SW_STRIDE + i] = wr[idx];
    }
    __syncthreads();

    const int wave = threadIdx.x >> 5;
    const int lane = threadIdx.x & 31;
    const int tile = blockIdx.x * 8 + wave;        // 16-node tile id
    const int n_tiles = N_NODES / 16;              // 625 (exact)
    if (tile >= n_tiles) return;                   // whole-wave guard: EXEC all-1 in WMMA

    const int m0 = tile * 16;
    const int arow = m0 + (lane & 15);             // A: M = lane%16
    const int koff = (lane >> 4) * 2;              // lanes 16-31 hold K=2,3
    const int bcol = lane & 15;                    // B: N = lane%16

    v8f acc0 = {}, acc1 = {}, acc2 = {}, acc3 = {};

    for (int k0 = 0; k0 < IN_F; k0 += 4) {
        v2f a = *(const v2f*)(feat_n + (size_t)arow * IN_F + k0 + koff);
        // B[k][n] = W_r[n][k] : read transposed from LDS
        v2f b0 = *(const v2f*)(sW + ( 0 + bcol) * SW_STRIDE + k0 + koff);
        v2f b1 = *(const v2f*)(sW + (16 + bcol) * SW_STRIDE + k0 + koff);
        v2f b2 = *(const v2f*)(sW + (32 + bcol) * SW_STRIDE + k0 + koff);
        v2f b3 = *(const v2f*)(sW + (48 + bcol) * SW_STRIDE + k0 + koff);
        acc0 = __builtin_amdgcn_wmma_f32_16x16x4_f32(false, a, false, b0, (short)0, acc0, false, false);
        acc1 = __builtin_amdgcn_wmma_f32_16x16x4_f32(false, a, false, b1, (short)0, acc1, false, false);
        acc2 = __builtin_amdgcn_wmma_f32_16x16x4_f32(false, a, false, b2, (short)0, acc2, false, false);
        acc3 = __builtin_amdgcn_wmma_f32_16x16x4_f32(false, a, false, b3, (short)0, acc3, false, false);
    }

    // D layout: VGPR j -> M=j (lanes 0-15) / M=j+8 (lanes 16-31), N = lane%16
    float* Hr = H + ((size_t)r * N_NODES + m0) * OUT_F;
    const int mAdd = (lane >> 4) * 8;
    const int nlan = lane & 15;
#pragma unroll
    for (int j = 0; j < 8; ++j) {
        float* row = Hr + (size_t)(mAdd + j) * OUT_F + nlan;
        row[0]  = acc0[j];
        row[16] = acc1[j];
        row[32] = acc2[j];
        row[48] = acc3[j];
    }
}

// ------------------------------------------------- edge gather + scatter-sum
// out[dst[e], :] += H[order[e], src[e], :]
__global__ __launch_bounds__(256) void scatter_kernel(const int* __restrict__ src,
                                                      const int* __restrict__ dst,
                                                      const int* __restrict__ order,
                                                      const float* __restrict__ H,
                                                      float* __restrict__ out) {
    int t = blockIdx.x * blockDim.x + threadIdx.x;   // E * 16 threads
    int e = t >> 4;
    if (e >= N_EDGES) return;
    int q = (t & 15) * 4;                            // 4 contiguous feats / thread
    int r = order[e];
    const float* h = H + (((size_t)r * N_NODES + src[e]) * OUT_F) + q;
    float4 v = *(const float4*)h;
    float* o = out + (size_t)dst[e] * OUT_F + q;
    atomicAdd(o + 0, v.x);
    atomicAdd(o + 1, v.y);
    atomicAdd(o + 2, v.z);
    atomicAdd(o + 3, v.w);
}

// ----------------------------------------- out = out * in_deg^-1/2 + bias
__global__ __launch_bounds__(256) void postnorm_kernel(float* __restrict__ out,
                                                       const float* __restrict__ deg_in,
                                                       const float* __restrict__ bias,
                                                       int n_elems) {
    int t = blockIdx.x * blockDim.x + threadIdx.x;
    if (t >= n_elems) return;
    int node = t >> 6;
    int o = t & 63;
    float s = rsqrtf(fmaxf(deg_in[node], 1.0f));
    out[t] = out[t] * s + bias[o];
}

extern "C" void kernel_launch(void* const* d_in, const int* in_sizes, int n_in,
                              void* d_out, int out_size, void* d_ws, size_t ws_size,
                              hipStream_t stream) {
    (void)in_sizes; (void)n_in; (void)out_size; (void)ws_size;

    const float* feat = (const float*)d_in[0];   // [N, 64]
    const int*   src  = (const int*)d_in[1];     // [E]
    const int*   dst  = (const int*)d_in[2];     // [E]
    const int*   ord  = (const int*)d_in[3];     // [E]
    const float* emb  = (const float*)d_in[4];   // [10, 4096]
    const float* bias = (const float*)d_in[5];   // [64]
    float* out = (float*)d_out;                  // [N, 64]

    // workspace layout (floats)
    float* ws       = (float*)d_ws;
    float* deg_out  = ws;                              // N
    float* deg_in   = ws + N_NODES;                    // N
    float* feat_n   = ws + 2 * N_NODES;                // N*64
    float* H        = ws + 2 * N_NODES + N_NODES * IN_F;  // 10*N*64

    const int NE = N_NODES * OUT_F;                    // 640000

    // 1) zero degrees + output
    zero_kernel<<<(2 * N_NODES + 255) / 256, 256, 0, stream>>>(deg_out, 2 * N_NODES);
    zero_kernel<<<(NE + 255) / 256, 256, 0, stream>>>(out, NE);

    // 2) degree counts
    degree_kernel<<<(N_EDGES + 255) / 256, 256, 0, stream>>>(src, dst, deg_out, deg_in, N_EDGES);

    // 3) pre-normalize source features
    prenorm_kernel<<<(NE + 255) / 256, 256, 0, stream>>>(feat, deg_out, feat_n, NE);

    // 4) per-relation dense GEMM on WMMA: H[r] = feat_n @ W_r^T
    dim3 ggrid((N_NODES / 16 + 7) / 8, N_REL, 1);      // (79, 10)
    rgcn_gemm_kernel<<<ggrid, 256, 0, stream>>>(feat_n, emb, H);

    // 5) gather + atomic scatter to destinations
    scatter_kernel<<<(N_EDGES * 16 + 255) / 256, 256, 0, stream>>>(src, dst, ord, H, out);

    // 6) post-normalize + bias
    postnorm_kernel<<<(NE + 255) / 256, 256, 0, stream>>>(out, deg_in, bias, NE);
}